// GraphClassifier_34617436406345
// MI455X (gfx1250) — compile-verified
//
#include <hip/hip_runtime.h>

#define N_NODES   50000
#define N_EDGES   800000
#define IN_DIM    64
#define HID       64
#define N_CLASSES 10
#define NUM_GRAPHS 512
#define BN_EPS    1e-5f
#define N_TILES   (N_NODES / 16)   // 3125, exact

typedef __attribute__((ext_vector_type(2))) float v2f;
typedef __attribute__((ext_vector_type(8))) float v8f;

// ---------------------------------------------------------------- utilities

__global__ void fill0_kernel(float* __restrict__ p, int n) {
    int i = blockIdx.x * blockDim.x + threadIdx.x;
    if (i < n) p[i] = 0.0f;
}

__global__ void deg_kernel(const int* __restrict__ dst, float* __restrict__ deg) {
    int e = blockIdx.x * blockDim.x + threadIdx.x;
    if (e < N_EDGES) atomicAdd(&deg[dst[e]], 1.0f);
}

__global__ void invdeg_kernel(const float* __restrict__ deg, float* __restrict__ inv) {
    int i = blockIdx.x * blockDim.x + threadIdx.x;
    if (i < N_NODES) inv[i] = 1.0f / fmaxf(deg[i], 1.0f);
}

// ------------------------------------------------- edge scatter-add (L2-bound)
// thread = (edge, 4-channel group); x rows are L2-resident (12.8 MB)

__global__ void aggregate_kernel(const float* __restrict__ x,
                                 const int* __restrict__ src,
                                 const int* __restrict__ dst,
                                 float* __restrict__ agg) {
    int idx = blockIdx.x * blockDim.x + threadIdx.x;
    int e = idx >> 4;
    if (e >= N_EDGES) return;
    int c4 = (idx & 15) << 2;
    int s = src[e];
    int d = dst[e];
    const float4 v = *(const float4*)(x + (size_t)s * HID + c4);
    float* a = agg + (size_t)d * HID + c4;
    atomicAdd(a + 0, v.x);
    atomicAdd(a + 1, v.y);
    atomicAdd(a + 2, v.z);
    atomicAdd(a + 3, v.w);
}

// ------------------------------------- fused dual-GEMM SAGE layer (f32 WMMA)
// out = relu(X@Ws + (inv_deg * Agg)@Wn + b)
// one wave -> 16-node x 64-channel tile; 128x v_wmma_f32_16x16x4_f32 per wave.
// Weights staged into LDS PAIR-SWIZZLED: pair k2=k/2 of rows (2k2, 2k2+1) is
// stored contiguously per column, so each lane's B fragment is ONE aligned
// ds_load_b64 directly into an even VGPR pair (no repacking movs).

__global__ __launch_bounds__(256) void sage_gemm_kernel(
    const float* __restrict__ X, const float* __restrict__ Agg,
    const float* __restrict__ invdeg,
    const float* __restrict__ Ws, const float* __restrict__ Wn,
    const float* __restrict__ bias, float* __restrict__ Out) {

    __shared__ float lws[HID * HID];   // pair-swizzled Ws
    __shared__ float lwn[HID * HID];   // pair-swizzled Wn
    __shared__ float lb[HID];

    const int tid = threadIdx.x;
    #pragma unroll
    for (int i = 0; i < 16; ++i) {                 // 4096 / 256 threads
        const int idx = tid + i * 256;             // source index = k*64 + n
        const int k = idx >> 6;
        const int n = idx & 63;
        const int didx = (((k >> 1) * 64 + n) << 1) | (k & 1);
        lws[didx] = Ws[idx];
        lwn[didx] = Wn[idx];
    }
    if (tid < HID) lb[tid] = bias[tid];
    __syncthreads();

    const int tile = blockIdx.x * 8 + (tid >> 5);
    if (tile >= N_TILES) return;                   // wave-uniform: EXEC all-1s

    const int lane = tid & 31;
    const int half = lane >> 4;                    // lanes 16-31 carry K+2 / M+8
    const int col  = lane & 15;
    const int rowbase = tile * 16;

    const float* __restrict__ xrow = X   + (size_t)(rowbase + col) * HID;
    const float* __restrict__ arow = Agg + (size_t)(rowbase + col) * HID;
    const float sc = invdeg[rowbase + col];        // fold mean-aggregation scale

    v8f acc[4];
    #pragma unroll
    for (int t = 0; t < 4; ++t) acc[t] = (v8f){0, 0, 0, 0, 0, 0, 0, 0};

    // ---- self path: X @ Ws
    #pragma unroll
    for (int k = 0; k < HID; k += 4) {
        const int kk = k + 2 * half;               // even
        const int k2 = (k >> 1) + half;            // pair index
        v2f a; a.x = xrow[kk]; a.y = xrow[kk + 1];
        #pragma unroll
        for (int t = 0; t < 4; ++t) {
            const v2f b = *(const v2f*)(lws + ((k2 * 64 + t * 16 + col) << 1));
            acc[t] = __builtin_amdgcn_wmma_f32_16x16x4_f32(
                false, a, false, b, (short)0, acc[t], false, false);
        }
    }

    // ---- neighbor path: (inv_deg * Agg) @ Wn
    #pragma unroll
    for (int k = 0; k < HID; k += 4) {
        const int kk = k + 2 * half;
        const int k2 = (k >> 1) + half;
        v2f a; a.x = arow[kk] * sc; a.y = arow[kk + 1] * sc;
        #pragma unroll
        for (int t = 0; t < 4; ++t) {
            const v2f b = *(const v2f*)(lwn + ((k2 * 64 + t * 16 + col) << 1));
            acc[t] = __builtin_amdgcn_wmma_f32_16x16x4_f32(
                false, a, false, b, (short)0, acc[t], false, false);
        }
    }

    // ---- bias + ReLU epilogue (C/D layout: VGPR j -> M=j+8*half, N=col)
    #pragma unroll
    for (int t = 0; t < 4; ++t) {
        const int n = t * 16 + col;
        const float bv = lb[n];
        #pragma unroll
        for (int j = 0; j < 8; ++j) {
            const float v = acc[t][j] + bv;
            Out[(size_t)(rowbase + j + 8 * half) * HID + n] = fmaxf(v, 0.0f);
        }
    }
}

// ------------------------------------------------------- BatchNorm (train mode)

__global__ void bn_stats_kernel(const float* __restrict__ h,
                                float* __restrict__ gsum,
                                float* __restrict__ gsq) {
    __shared__ float ssum[256];
    __shared__ float ssq[256];
    const int tid = threadIdx.x;
    const int c  = tid & 63;
    const int rb = tid >> 6;
    float sum = 0.0f, sq = 0.0f;
    for (int i = blockIdx.x * 4 + rb; i < N_NODES; i += gridDim.x * 4) {
        const float v = h[(size_t)i * HID + c];
        sum += v;
        sq  += v * v;
    }
    ssum[tid] = sum;
    ssq[tid]  = sq;
    __syncthreads();
    if (tid < 64) {
        sum = ssum[tid] + ssum[tid + 64] + ssum[tid + 128] + ssum[tid + 192];
        sq  = ssq[tid]  + ssq[tid + 64]  + ssq[tid + 128]  + ssq[tid + 192];
        atomicAdd(&gsum[c], sum);
        atomicAdd(&gsq[c],  sq);
    }
}

__global__ void bn_finalize_kernel(const float* __restrict__ gsum,
                                   const float* __restrict__ gsq,
                                   const float* __restrict__ g,
                                   const float* __restrict__ be,
                                   float* __restrict__ scale,
                                   float* __restrict__ shift) {
    const int c = threadIdx.x;
    if (c < HID) {
        const float inv_n = 1.0f / (float)N_NODES;
        const float mean = gsum[c] * inv_n;
        const float var  = gsq[c] * inv_n - mean * mean;   // population var (ddof=0)
        const float s = g[c] * rsqrtf(var + BN_EPS);
        scale[c] = s;
        shift[c] = be[c] - mean * s;
    }
}

__global__ void bn_apply_kernel(float* __restrict__ h,
                                const float* __restrict__ scale,
                                const float* __restrict__ shift) {
    const int idx = blockIdx.x * blockDim.x + threadIdx.x;
    if (idx < N_NODES * HID) {
        const int c = idx & 63;
        h[idx] = h[idx] * scale[c] + shift[c];
    }
}

// ---------------------------------------------------------------- sum pooling

__global__ void pool_kernel(const float* __restrict__ h,
                            const int* __restrict__ gi,
                            float* __restrict__ pooled) {
    const int idx = blockIdx.x * blockDim.x + threadIdx.x;
    const int n = idx >> 4;
    if (n >= N_NODES) return;
    const int c4 = (idx & 15) << 2;
    const int g = gi[n];
    const float4 v = *(const float4*)(h + (size_t)n * HID + c4);
    float* p = pooled + (size_t)g * HID + c4;
    atomicAdd(p + 0, v.x);
    atomicAdd(p + 1, v.y);
    atomicAdd(p + 2, v.z);
    atomicAdd(p + 3, v.w);
}

// ---------------------------------------------------------------- MLP head

__global__ void fc_kernel(const float* __restrict__ in, const float* __restrict__ W,
                          const float* __restrict__ b, float* __restrict__ out,
                          int R, int K, int Nout, int do_relu) {
    const int idx = blockIdx.x * blockDim.x + threadIdx.x;
    if (idx >= R * Nout) return;
    const int r = idx / Nout;
    const int n = idx - r * Nout;
    float s = b[n];
    for (int k = 0; k < K; ++k)
        s = fmaf(in[(size_t)r * K + k], W[(size_t)k * Nout + n], s);
    out[idx] = do_relu ? fmaxf(s, 0.0f) : s;
}

// ---------------------------------------------------------------- launcher

extern "C" void kernel_launch(void* const* d_in, const int* in_sizes, int n_in,
                              void* d_out, int out_size, void* d_ws, size_t ws_size,
                              hipStream_t stream) {
    const float* X    = (const float*)d_in[0];
    const int*   esrc = (const int*)  d_in[1];
    const int*   edst = (const int*)  d_in[2];
    const int*   gind = (const int*)  d_in[3];
    // d_in[4] = labels (unused by forward)
    const float* Ws1 = (const float*)d_in[5];
    const float* Wn1 = (const float*)d_in[6];
    const float* b1  = (const float*)d_in[7];
    const float* Ws2 = (const float*)d_in[8];
    const float* Wn2 = (const float*)d_in[9];
    const float* b2  = (const float*)d_in[10];
    const float* Ws3 = (const float*)d_in[11];
    const float* Wn3 = (const float*)d_in[12];
    const float* b3  = (const float*)d_in[13];
    const float* g2  = (const float*)d_in[14];
    const float* be2 = (const float*)d_in[15];
    const float* g3  = (const float*)d_in[16];
    const float* be3 = (const float*)d_in[17];
    const float* fcW1 = (const float*)d_in[18];
    const float* fcb1 = (const float*)d_in[19];
    const float* fcW2 = (const float*)d_in[20];
    const float* fcb2 = (const float*)d_in[21];
    const float* fcW3 = (const float*)d_in[22];
    const float* fcb3 = (const float*)d_in[23];

    // workspace layout (floats); ~40 MB total
    float* ws      = (float*)d_ws;
    float* agg     = ws;                       // 3,200,000
    float* hA      = agg     + (size_t)N_NODES * HID;
    float* hB      = hA      + (size_t)N_NODES * HID;
    float* deg     = hB      + (size_t)N_NODES * HID;   // 50,000
    float* invd    = deg     + N_NODES;                 // 50,000
    float* bnsum   = invd    + N_NODES;                 // 64
    float* bnsq    = bnsum   + HID;                     // 64
    float* bnscale = bnsq    + HID;                     // 64
    float* bnshift = bnscale + HID;                     // 64
    float* pooled  = bnshift + HID;                     // 32,768
    float* t1      = pooled  + NUM_GRAPHS * HID;        // 65,536
    float* t2      = t1      + NUM_GRAPHS * 2 * HID;    // 32,768

    auto Z = [&](float* p, int n) {
        fill0_kernel<<<(n + 255) / 256, 256, 0, stream>>>(p, n);
    };

    auto sage = [&](const float* xin, const float* Wsp, const float* Wnp,
                    const float* bp, float* hout) {
        Z(agg, N_NODES * HID);
        aggregate_kernel<<<(N_EDGES * 16 + 255) / 256, 256, 0, stream>>>(xin, esrc, edst, agg);
        sage_gemm_kernel<<<(N_TILES + 7) / 8, 256, 0, stream>>>(xin, agg, invd, Wsp, Wnp, bp, hout);
    };

    auto bn = [&](float* h, const float* g, const float* be) {
        Z(bnsum, HID);
        Z(bnsq, HID);
        bn_stats_kernel<<<256, 256, 0, stream>>>(h, bnsum, bnsq);
        bn_finalize_kernel<<<1, 64, 0, stream>>>(bnsum, bnsq, g, be, bnscale, bnshift);
        bn_apply_kernel<<<(N_NODES * HID + 255) / 256, 256, 0, stream>>>(h, bnscale, bnshift);
    };

    // degree / mean-aggregation scale (shared by all 3 layers)
    Z(deg, N_NODES);
    deg_kernel<<<(N_EDGES + 255) / 256, 256, 0, stream>>>(edst, deg);
    invdeg_kernel<<<(N_NODES + 255) / 256, 256, 0, stream>>>(deg, invd);

    // 3x (SAGE -> BN) with BN skipped after layer 3, per reference
    sage(X,  Ws1, Wn1, b1, hA);
    bn(hA, g2, be2);
    sage(hA, Ws2, Wn2, b2, hB);
    bn(hB, g3, be3);
    sage(hB, Ws3, Wn3, b3, hA);

    // per-graph sum pooling
    Z(pooled, NUM_GRAPHS * HID);
    pool_kernel<<<(N_NODES * 16 + 255) / 256, 256, 0, stream>>>(hA, gind, pooled);

    // MLP head -> logits [512, 10]
    fc_kernel<<<(NUM_GRAPHS * 2 * HID + 255) / 256, 256, 0, stream>>>(
        pooled, fcW1, fcb1, t1, NUM_GRAPHS, HID, 2 * HID, 1);
    fc_kernel<<<(NUM_GRAPHS * HID + 255) / 256, 256, 0, stream>>>(
        t1, fcW2, fcb2, t2, NUM_GRAPHS, 2 * HID, HID, 1);
    fc_kernel<<<(NUM_GRAPHS * N_CLASSES + 255) / 256, 256, 0, stream>>>(
        t2, fcW3, fcb3, (float*)d_out, NUM_GRAPHS, HID, N_CLASSES, 0);
}